// HeterogeneousGNNEncoder_850403524849
// MI455X (gfx1250) — compile-verified
//
#include <hip/hip_runtime.h>
#include <hip/hip_bf16.h>

// ---------------------------------------------------------------------------
// Heterogeneous GAT encoder for gfx1250 (MI455X).
// Dense per-relation linears run on the fp32 WMMA path (v_wmma_f32_16x16x4_f32);
// attention-dst scores are algebraically folded into a [128,4] matrix so only
// one 128x128 GEMM per relation is needed. Edge softmax/scatter uses native
// fp32 global atomics; LayerNorm is one wave32 per node.
// ---------------------------------------------------------------------------

typedef float v2f __attribute__((ext_vector_type(2)));
typedef float v8f __attribute__((ext_vector_type(8)));

#define CDIV(a, b) (((a) + (b) - 1) / (b))

static __device__ __forceinline__ unsigned fenc(float f) {
  unsigned u = __float_as_uint(f);
  return (u & 0x80000000u) ? ~u : (u | 0x80000000u);
}
static __device__ __forceinline__ float fdec(unsigned u) {
  unsigned b = (u & 0x80000000u) ? (u & 0x7FFFFFFFu) : ~u;
  return __uint_as_float(b);
}
static __device__ __forceinline__ void atomAddF(float* p, float v) {
  __hip_atomic_fetch_add(p, v, __ATOMIC_RELAXED, __HIP_MEMORY_SCOPE_AGENT);
}

// ---------------- WMMA GEMM: C[N,128] = A[N,K] @ W[K,128] (+bias) ----------
// One wave computes a 16-row x 128-col slab with 8 f32 16x16x4 accumulators.
// W is staged in LDS as K-pairs: pair p holds rows 2p,2p+1 interleaved per
// column (lw[p*288 + 2*col + parity]), so each B fragment is one aligned
// ds_load_b64 straight into an even VGPR pair (no repack movs). The 288-float
// pair pitch puts the two half-wave read windows 32 banks apart: conflict-free.
// A row indices are clamped (stores masked), and the K loop is split into an
// unguarded main body plus one guarded ragged-tail step, so the hot loop has
// no exec-mask branches.
__launch_bounds__(256)
__global__ void k_gemm_wmma(const float* __restrict__ A, const float* __restrict__ W,
                            const float* __restrict__ bias, float* __restrict__ C,
                            int N, int K) {
  __shared__ float lw[64 * 288];
  const int tid = threadIdx.x;
  const int kpad = (K + 3) & ~3;
  for (int i = tid; i < kpad * 128; i += 256) {
    int r = i >> 7, c = i & 127;
    lw[(r >> 1) * 288 + 2 * c + (r & 1)] = (r < K) ? W[r * 128 + c] : 0.0f;
  }
  __syncthreads();
  const int wave = tid >> 5, lane = tid & 31;
  const int half = lane >> 4, l16 = lane & 15;
  const int row0 = (blockIdx.x * 8 + wave) * 16;
  int arow = row0 + l16;
  if (arow >= N) arow = N - 1;  // clamp; OOB rows compute garbage, stores masked
  const float* Arow = A + (long)arow * K;
  v8f acc[8] = {};
  const int kmain = K & ~3;
  for (int k = 0; k < kmain; k += 4) {
    const int kk = k + half * 2;  // lanes 0-15: K=k,k+1 ; lanes 16-31: K=k+2,k+3
    v2f a;
    a.x = Arow[kk];
    a.y = Arow[kk + 1];
    const float* bbase = &lw[(size_t)((k >> 1) + half) * 288 + 2 * l16];
#pragma unroll
    for (int t = 0; t < 8; ++t) {
      v2f b = *(const v2f*)(bbase + 32 * t);
      acc[t] = __builtin_amdgcn_wmma_f32_16x16x4_f32(false, a, false, b, (short)0,
                                                     acc[t], false, false);
    }
  }
  if (kmain < K) {  // ragged tail (input projections: K = 10/7/3)
    const int kk = kmain + half * 2;
    v2f a;
    a.x = (kk < K) ? Arow[kk] : 0.0f;
    a.y = (kk + 1 < K) ? Arow[kk + 1] : 0.0f;
    const float* bbase = &lw[(size_t)((kmain >> 1) + half) * 288 + 2 * l16];
#pragma unroll
    for (int t = 0; t < 8; ++t) {
      v2f b = *(const v2f*)(bbase + 32 * t);
      acc[t] = __builtin_amdgcn_wmma_f32_16x16x4_f32(false, a, false, b, (short)0,
                                                     acc[t], false, false);
    }
  }
#pragma unroll
  for (int t = 0; t < 8; ++t) {
#pragma unroll
    for (int v = 0; v < 8; ++v) {
      int m = row0 + half * 8 + v;  // C/D layout: VGPR v holds M=v (+8 for hi half)
      if (m < N) {
        int c = t * 16 + l16;
        float val = acc[t][v];
        if (bias) val += bias[c];
        C[(long)m * 128 + c] = val;
      }
    }
  }
}

// Wd4[r,h] = sum_c W[r, h*32+c] * att_dst[h,c]   (folds att_dst into W)
__global__ void k_wd4(const float* __restrict__ W, const float* __restrict__ attd,
                      float* __restrict__ wd4) {
  int t = blockIdx.x * blockDim.x + threadIdx.x;
  if (t >= 512) return;
  int r = t >> 2, h = t & 3;
  const float* wr = W + r * 128 + h * 32;
  const float* av = attd + h * 32;
  float s = 0.f;
#pragma unroll 8
  for (int c = 0; c < 32; ++c) s += wr[c] * av[c];
  wd4[t] = s;
}

// as[i,h] = sum_c hs[i, h*32+c] * att_src[h,c]
__global__ void k_score_src(const float* __restrict__ hs, const float* __restrict__ atts,
                            float* __restrict__ as_, int N) {
  int t = blockIdx.x * blockDim.x + threadIdx.x;
  if (t >= N * 4) return;
  int i = t >> 2, h = t & 3;
  const float* p = hs + (long)i * 128 + h * 32;
  const float* a = atts + h * 32;
  float s = 0.f;
#pragma unroll 8
  for (int c = 0; c < 32; ++c) s += p[c] * a[c];
  as_[t] = s;
}

// ad[i,h] = h_dst[i,:] . Wd4[:,h]
__global__ void k_score_dst(const float* __restrict__ hd, const float* __restrict__ wd4,
                            float* __restrict__ ad_, int N) {
  int t = blockIdx.x * blockDim.x + threadIdx.x;
  if (t >= N * 4) return;
  int i = t >> 2, h = t & 3;
  const float* p = hd + (long)i * 128;
  float s = 0.f;
#pragma unroll 8
  for (int r = 0; r < 128; ++r) s += p[r] * wd4[r * 4 + h];
  ad_[t] = s;
}

// pass 1: leaky-relu logits + segment max (ordered-uint atomicMax)
__global__ void k_edge_logit(const int* __restrict__ si, const int* __restrict__ di,
                             const float* __restrict__ as_, const float* __restrict__ ad_,
                             float* __restrict__ ae, unsigned* __restrict__ menc, int E) {
  int t = blockIdx.x * blockDim.x + threadIdx.x;
  if (t >= E * 4) return;
  int e = t >> 2, h = t & 3;
  float a = as_[si[e] * 4 + h] + ad_[di[e] * 4 + h];
  a = (a >= 0.f) ? a : 0.2f * a;
  ae[t] = a;
  atomicMax(&menc[di[e] * 4 + h], fenc(a));
}

// pass 2: exp(a - max) in place + segment sum
__global__ void k_edge_exp(const int* __restrict__ di, float* __restrict__ ae,
                           const unsigned* __restrict__ menc, float* __restrict__ denom,
                           int E) {
  int t = blockIdx.x * blockDim.x + threadIdx.x;
  if (t >= E * 4) return;
  int e = t >> 2, h = t & 3;
  float ex = __expf(ae[t] - fdec(menc[di[e] * 4 + h]));
  ae[t] = ex;
  atomAddF(&denom[di[e] * 4 + h], ex);
}

// pass 3: alpha-weighted message scatter (float4 gathers, 32 fp32 atomics)
__global__ void k_edge_scatter(const int* __restrict__ si, const int* __restrict__ di,
                               const float* __restrict__ ex, const float* __restrict__ denom,
                               const float* __restrict__ hs, float* __restrict__ accD, int E) {
  int t = blockIdx.x * blockDim.x + threadIdx.x;
  if (t >= E * 4) return;
  int e = t >> 2, h = t & 3;
  int sv = si[e], dv = di[e];
  float alpha = ex[t] / (denom[dv * 4 + h] + 1e-16f);
  const float4* hp = (const float4*)(hs + (long)sv * 128 + h * 32);
  float* ap = accD + (long)dv * 128 + h * 32;
#pragma unroll
  for (int q = 0; q < 8; ++q) {
    float4 v = hp[q];
    atomAddF(ap + q * 4 + 0, alpha * v.x);
    atomAddF(ap + q * 4 + 1, alpha * v.y);
    atomAddF(ap + q * 4 + 2, alpha * v.z);
    atomAddF(ap + q * 4 + 3, alpha * v.w);
  }
}

__global__ void k_add_bias(float* __restrict__ acc, const float* __restrict__ bias, int N) {
  int t = blockIdx.x * blockDim.x + threadIdx.x;
  if (t >= N * 128) return;
  acc[t] += bias[t & 127];
}

// mean-over-relations -> relu -> +residual -> LayerNorm ; one wave32 per node
__launch_bounds__(256)
__global__ void k_finalize_ln(float* __restrict__ h, const float* __restrict__ acc,
                              float inv_nrel, const float* __restrict__ g,
                              const float* __restrict__ b, int N) {
  int wave = threadIdx.x >> 5, lane = threadIdx.x & 31;
  int node = blockIdx.x * 8 + wave;
  if (node >= N) return;
  long base = (long)node * 128;
  float v[4];
  float s = 0.f;
#pragma unroll
  for (int j = 0; j < 4; ++j) {
    float nv = acc[base + lane * 4 + j] * inv_nrel;
    nv = fmaxf(nv, 0.f);
    v[j] = nv + h[base + lane * 4 + j];
    s += v[j];
  }
#pragma unroll
  for (int o = 16; o > 0; o >>= 1) s += __shfl_xor(s, o, 32);
  float mu = s * (1.0f / 128.0f);
  float var = 0.f;
#pragma unroll
  for (int j = 0; j < 4; ++j) { float d = v[j] - mu; var += d * d; }
#pragma unroll
  for (int o = 16; o > 0; o >>= 1) var += __shfl_xor(var, o, 32);
  float rstd = rsqrtf(var * (1.0f / 128.0f) + 1e-5f);
#pragma unroll
  for (int j = 0; j < 4; ++j) {
    int c = lane * 4 + j;
    h[base + c] = (v[j] - mu) * rstd * g[c] + b[c];
  }
}

__global__ void k_pool_acc(const float* __restrict__ h, const int* __restrict__ batch,
                           float* __restrict__ pooled, int coff, int N) {
  int t = blockIdx.x * blockDim.x + threadIdx.x;
  if (t >= N * 128) return;
  int i = t >> 7, c = t & 127;
  atomAddF(&pooled[(long)batch[i] * 512 + coff + c], h[t]);
}

__global__ void k_pool_cnt(const int* __restrict__ batch, float* __restrict__ cnt, int N) {
  int t = blockIdx.x * blockDim.x + threadIdx.x;
  if (t >= N) return;
  atomAddF(&cnt[batch[t]], 1.0f);
}

__global__ void k_pool_fin(const float* __restrict__ pooled, const float* __restrict__ cnt,
                           float* __restrict__ out) {
  int t = blockIdx.x * blockDim.x + threadIdx.x;
  if (t >= 32 * 512) return;
  int g = t >> 9, col = t & 511, nt = col >> 7;
  out[t] = pooled[t] / fmaxf(cnt[nt * 32 + g], 1.0f);
}

// ---------------------------------------------------------------------------
// Input flattening assumption (JAX pytree: dict keys sorted within each entry,
// top-level entries in reference-signature order):
//  0..3   x_block, x_spmt, x_crane, x_facility
//  4+L*34+r*4 .. +3   layer L, relation r (alphabetical): W, att_dst, att_src, bias
//  4+L*34+32/33       layer L: ln_b, ln_g
//  72..79 proj (block, crane, facility, spmt): W, b each
//  80..95 edges (alphabetical relation order): si, di pairs
//  96..99 batch: block, crane, facility, spmt
// Relation alphabetical order:
//  0 block__at__facility  1 block__needs_lift__crane  2 block__needs_transport__spmt
//  3 block__precedes__block  4 crane__at__facility  5 crane__can_lift__block
//  6 spmt__at__facility  7 spmt__can_transport__block
// ---------------------------------------------------------------------------

extern "C" void kernel_launch(void* const* d_in, const int* in_sizes, int n_in,
                              void* d_out, int out_size, void* d_ws, size_t ws_size,
                              hipStream_t stream) {
  (void)in_sizes; (void)n_in; (void)out_size; (void)ws_size;

  static const int NN[4]   = {100000, 10000, 5000, 1000};  // block, spmt, crane, facility
  static const int DD[4]   = {8, 10, 7, 3};
  static const int NREL[4] = {3, 1, 1, 3};
  static const int POFF[4] = {0, 128, 256, 384};
  static const int projW_i[4] = {72, 78, 74, 76};
  static const int batch_i[4] = {96, 99, 97, 98};
  struct RelT { int s, d, e; };
  static const RelT R[8] = {
      {0, 3, 100000}, {0, 2, 200000}, {0, 1, 400000}, {0, 0, 200000},
      {2, 3, 5000},   {2, 0, 200000}, {1, 3, 10000},  {1, 0, 400000}};

  char* ws = (char*)d_ws;
  size_t off = 0;
  auto carve = [&](size_t bytes) -> char* {
    char* p = ws + off;
    off = (off + bytes + 255) & ~(size_t)255;
    return p;
  };
  float* h[4];
  float* acc[4];
  for (int nt = 0; nt < 4; ++nt) h[nt] = (float*)carve((size_t)NN[nt] * 128 * 4);
  for (int nt = 0; nt < 4; ++nt) acc[nt] = (float*)carve((size_t)NN[nt] * 128 * 4);
  float*    hs     = (float*)carve((size_t)100000 * 128 * 4);
  float*    asb    = (float*)carve((size_t)100000 * 4 * 4);
  float*    adb    = (float*)carve((size_t)100000 * 4 * 4);
  unsigned* menc   = (unsigned*)carve((size_t)100000 * 4 * 4);
  float*    denom  = (float*)carve((size_t)100000 * 4 * 4);
  float*    ae     = (float*)carve((size_t)400000 * 4 * 4);
  float*    wd4    = (float*)carve(512 * 4);
  float*    pooled = (float*)carve(32 * 512 * 4);
  float*    cnt    = (float*)carve(4 * 32 * 4);

  // Input projections: h[nt] = x_nt @ W_proj + b  (ragged K handled by tail step)
  for (int nt = 0; nt < 4; ++nt) {
    k_gemm_wmma<<<CDIV(NN[nt], 128), 256, 0, stream>>>(
        (const float*)d_in[nt], (const float*)d_in[projW_i[nt]],
        (const float*)d_in[projW_i[nt] + 1], h[nt], NN[nt], DD[nt]);
  }

  for (int L = 0; L < 2; ++L) {
    const int base = 4 + L * 34;
    for (int nt = 0; nt < 4; ++nt)
      hipMemsetAsync(acc[nt], 0, (size_t)NN[nt] * 128 * 4, stream);
    for (int r = 0; r < 8; ++r) {
      const float* W    = (const float*)d_in[base + r * 4 + 0];
      const float* attd = (const float*)d_in[base + r * 4 + 1];
      const float* atts = (const float*)d_in[base + r * 4 + 2];
      const float* bias = (const float*)d_in[base + r * 4 + 3];
      const int*   si   = (const int*)d_in[80 + r * 2];
      const int*   di   = (const int*)d_in[81 + r * 2];
      const int s = R[r].s, d = R[r].d, E = R[r].e;
      const int Ns = NN[s], Nd = NN[d];
      k_wd4<<<2, 256, 0, stream>>>(W, attd, wd4);
      k_gemm_wmma<<<CDIV(Ns, 128), 256, 0, stream>>>(h[s], W, nullptr, hs, Ns, 128);
      k_score_src<<<CDIV(Ns * 4, 256), 256, 0, stream>>>(hs, atts, asb, Ns);
      k_score_dst<<<CDIV(Nd * 4, 256), 256, 0, stream>>>(h[d], wd4, adb, Nd);
      hipMemsetAsync(menc, 0, (size_t)Nd * 4 * 4, stream);
      hipMemsetAsync(denom, 0, (size_t)Nd * 4 * 4, stream);
      k_edge_logit<<<CDIV(E * 4, 256), 256, 0, stream>>>(si, di, asb, adb, ae, menc, E);
      k_edge_exp<<<CDIV(E * 4, 256), 256, 0, stream>>>(di, ae, menc, denom, E);
      k_edge_scatter<<<CDIV(E * 4, 256), 256, 0, stream>>>(si, di, ae, denom, hs, acc[d], E);
      k_add_bias<<<CDIV(Nd * 128, 256), 256, 0, stream>>>(acc[d], bias, Nd);
    }
    const float* lnb = (const float*)d_in[base + 32];
    const float* lng = (const float*)d_in[base + 33];
    for (int nt = 0; nt < 4; ++nt)
      k_finalize_ln<<<CDIV(NN[nt], 8), 256, 0, stream>>>(h[nt], acc[nt], 1.0f / NREL[nt],
                                                         lng, lnb, NN[nt]);
  }

  hipMemsetAsync(pooled, 0, 32 * 512 * 4, stream);
  hipMemsetAsync(cnt, 0, 4 * 32 * 4, stream);
  for (int nt = 0; nt < 4; ++nt) {
    const int* b = (const int*)d_in[batch_i[nt]];
    k_pool_acc<<<CDIV(NN[nt] * 128, 256), 256, 0, stream>>>(h[nt], b, pooled, POFF[nt], NN[nt]);
    k_pool_cnt<<<CDIV(NN[nt], 256), 256, 0, stream>>>(b, cnt + nt * 32, NN[nt]);
  }
  k_pool_fin<<<64, 256, 0, stream>>>(pooled, cnt, (float*)d_out);
}